// DetectionLayer_71459665870872
// MI455X (gfx1250) — compile-verified
//
#include <hip/hip_runtime.h>
#include <cstdint>

#define S 7
#define NUM 2
#define CLS 20
#define MAX_BOX 30
#define CH (NUM * 5 + CLS)   // 30 channels per cell
#define XLEN (S * S * CH)    // 1470 floats per example
#define NSLOT (S * S * NUM)  // 98 (cell,box) slots
#define NCELL (S * S)        // 49
#define NCLS (S * S * CLS)   // 980
#define COORD_SCALE 5.0f
#define NOOBJ_SCALE 0.5f

#if defined(__HIP_DEVICE_COMPILE__)
#if __has_builtin(__builtin_amdgcn_global_load_async_to_lds_b32) && \
    __has_builtin(__builtin_amdgcn_s_wait_asynccnt)
#define USE_ASYNC_LDS 1
#endif
#endif

#ifdef USE_ASYNC_LDS
typedef __attribute__((address_space(1))) int gint_t;  // global (device) AS
typedef __attribute__((address_space(3))) int lint_t;  // LDS AS (32-bit ptr)
#endif

__device__ __forceinline__ float sigmoidf_(float x) {
  return 1.0f / (1.0f + expf(-x));
}

__global__ __launch_bounds__(128) void yolo_loss_kernel(
    const float* __restrict__ x, const float* __restrict__ boxes,
    const int* __restrict__ labels, const int* __restrict__ nbox,
    const int* __restrict__ net_h, const int* __restrict__ net_w,
    float* __restrict__ per_loss) {
  __shared__ __align__(16) float xs[XLEN];
  __shared__ float loc_t[NSLOT][4];
  __shared__ float iou_t[NSLOT];
  __shared__ float ba[NSLOT];
  __shared__ float ca[NCELL];
  __shared__ float cls_t[NCLS];
  __shared__ float wsum[4];

  const int b = blockIdx.x;
  const int tid = threadIdx.x;
  const int bdim = blockDim.x;
  const float* xb = x + (size_t)b * XLEN;
  const float img_w = (float)net_w[0];
  const float img_h = (float)net_h[0];

  // ---- Stage predictions into LDS (CDNA5 async global->LDS path) ----
#ifdef USE_ASYNC_LDS
  for (int i = tid; i < XLEN; i += bdim) {
    uintptr_t ga = (uintptr_t)(xb + i);
    unsigned la = (unsigned)(uintptr_t)(&xs[i]);  // low 32 bits = LDS offset
    __builtin_amdgcn_global_load_async_to_lds_b32((gint_t*)ga, (lint_t*)la, 0,
                                                  0);
  }
#else
  for (int i = tid; i < XLEN; i += bdim) xs[i] = xb[i];
#endif

  // ---- Zero target tensors while async loads are in flight ----
  for (int i = tid; i < NSLOT; i += bdim) {
    loc_t[i][0] = 0.f; loc_t[i][1] = 0.f; loc_t[i][2] = 0.f; loc_t[i][3] = 0.f;
    iou_t[i] = 0.f;
    ba[i] = 0.f;
  }
  for (int i = tid; i < NCELL; i += bdim) ca[i] = 0.f;
  for (int i = tid; i < NCLS; i += bdim) cls_t[i] = 0.f;

#ifdef USE_ASYNC_LDS
  __builtin_amdgcn_s_wait_asynccnt(0);
#endif
  __syncthreads();

  // ---- Sequential GT assignment scan (last writer wins, as in lax.scan) ----
  if (tid == 0) {
    int nb = nbox[b];
    if (nb > MAX_BOX) nb = MAX_BOX;
    if (nb < 0) nb = 0;
    const float cw = img_w / (float)S;
    const float chh = img_h / (float)S;
    for (int k = 0; k < nb; ++k) {
      const float* bb = boxes + ((size_t)b * MAX_BOX + k) * 4;
      float x1 = bb[0], y1 = bb[1], x2 = bb[2], y2 = bb[3];
      float cx = (x1 + x2) * 0.5f, cy = (y1 + y2) * 0.5f;
      float w = x2 - x1, h = y2 - y1;
      int cj = (int)floorf(cx / cw);   // x index (column)
      int ci = (int)floorf(cy / chh);  // y index (row)
      cj = min(max(cj, 0), S - 1);
      ci = min(max(ci, 0), S - 1);
      float cxn = cx / img_w, cyn = cy / img_h;
      float wn = w / img_w, hn = h / img_h;
      float gx1 = cxn - wn * 0.5f, gy1 = cyn - hn * 0.5f;
      float gx2 = cxn + wn * 0.5f, gy2 = cyn + hn * 0.5f;
      float areaB = (gx2 - gx1) * (gy2 - gy1);
      int cell = ci * S + cj;
      float iouv[NUM];
      for (int n = 0; n < NUM; ++n) {
        int base = cell * CH + n * 5;
        float sx = sigmoidf_(xs[base + 0]);
        float sy = sigmoidf_(xs[base + 1]);
        float sw = sigmoidf_(xs[base + 2]);
        float sh = sigmoidf_(xs[base + 3]);
        float pcx = (sx + (float)cj) / (float)S;
        float pcy = (sy + (float)ci) / (float)S;
        float px1 = pcx - sw * 0.5f, py1 = pcy - sh * 0.5f;
        float px2 = pcx + sw * 0.5f, py2 = pcy + sh * 0.5f;
        float lx = fmaxf(px1, gx1), ly = fmaxf(py1, gy1);
        float rx = fminf(px2, gx2), ry = fminf(py2, gy2);
        float iw = fmaxf(rx - lx, 0.f), ih = fmaxf(ry - ly, 0.f);
        float inter = iw * ih;
        float areaA = (px2 - px1) * (py2 - py1);
        iouv[n] = inter / (areaA + areaB - inter);
      }
      int m = (iouv[1] > iouv[0]) ? 1 : 0;  // argmax, first index on tie
      int slot = cell * NUM + m;
      loc_t[slot][0] = cxn * (float)S - (float)cj;
      loc_t[slot][1] = cyn * (float)S - (float)ci;
      loc_t[slot][2] = wn;
      loc_t[slot][3] = hn;
      iou_t[slot] = iouv[m];
      ba[slot] = 1.f;
      ca[cell] = 1.f;
      int lab = labels[(size_t)b * MAX_BOX + k];
      lab = min(max(lab, 0), CLS - 1);
      cls_t[cell * CLS + lab] = 1.f;
    }
  }
  __syncthreads();

  // ---- Parallel loss terms ----
  float acc = 0.f;
  if (tid < NSLOT) {
    int slot = tid;
    int cell = slot / NUM;
    int n = slot % NUM;
    int base = cell * CH + n * 5;
    float sx = sigmoidf_(xs[base + 0]);
    float sy = sigmoidf_(xs[base + 1]);
    float sw = sigmoidf_(xs[base + 2]);
    float sh = sigmoidf_(xs[base + 3]);
    float sc = sigmoidf_(xs[base + 4]);
    float a = ba[slot];
    float dx = sx - loc_t[slot][0];
    float dy = sy - loc_t[slot][1];
    float dw = sqrtf(sw) - sqrtf(loc_t[slot][2]);
    float dh = sqrtf(sh) - sqrtf(loc_t[slot][3]);
    float coord = COORD_SCALE * (dx * dx + dy * dy + dw * dw + dh * dh);
    float dc = sc - iou_t[slot];
    float obj = dc * dc;
    float noobj = NOOBJ_SCALE * sc * sc;
    acc += a * (coord + obj) + (1.f - a) * noobj;
  }
  for (int idx = tid; idx < NCLS; idx += bdim) {
    int cell = idx / CLS;
    int c = idx % CLS;
    float v = sigmoidf_(xs[cell * CH + NUM * 5 + c]);
    float d = v - cls_t[idx];
    acc += ca[cell] * d * d;
  }

  // ---- Deterministic block reduction (wave32 shuffle tree + fixed order) ----
  for (int off = 16; off > 0; off >>= 1) acc += __shfl_down(acc, off, 32);
  int wave = tid >> 5;
  if ((tid & 31) == 0) wsum[wave] = acc;
  __syncthreads();
  if (tid == 0) {
    per_loss[b] = ((wsum[0] + wsum[1]) + wsum[2]) + wsum[3];
  }
}

__global__ __launch_bounds__(256) void reduce_mean_kernel(
    const float* __restrict__ per_loss, float* __restrict__ out, int n) {
  __shared__ float sm[256];
  int tid = threadIdx.x;
  float acc = 0.f;
  for (int i = tid; i < n; i += 256) acc += per_loss[i];
  sm[tid] = acc;
  __syncthreads();
  for (int s = 128; s > 0; s >>= 1) {
    if (tid < s) sm[tid] += sm[tid + s];
    __syncthreads();
  }
  if (tid == 0) out[0] = sm[0] / (float)n;
}

extern "C" void kernel_launch(void* const* d_in, const int* in_sizes, int n_in,
                              void* d_out, int out_size, void* d_ws,
                              size_t ws_size, hipStream_t stream) {
  const float* b_x = (const float*)d_in[0];
  const float* b_boxes = (const float*)d_in[1];
  const int* b_labels = (const int*)d_in[2];
  const int* b_nbox = (const int*)d_in[3];
  const int* net_h = (const int*)d_in[4];
  const int* net_w = (const int*)d_in[5];

  const int B = in_sizes[0] / XLEN;
  float* per_loss = (float*)d_ws;  // B floats of scratch

  yolo_loss_kernel<<<B, 128, 0, stream>>>(b_x, b_boxes, b_labels, b_nbox,
                                          net_h, net_w, per_loss);
  reduce_mean_kernel<<<1, 256, 0, stream>>>(per_loss, (float*)d_out, B);
}